// GCN_47991964565963
// MI455X (gfx1250) — compile-verified
//
#include <hip/hip_runtime.h>

typedef __attribute__((ext_vector_type(2))) float v2f;
typedef __attribute__((ext_vector_type(8))) float v8f;

#define NFEAT 128
#define NHID  128

// ---------------------------------------------------------------------------
// deg[i] = 1.0f  (self-loop contribution)
__global__ void gcn_init_deg(float* __restrict__ deg, int n) {
    int i = blockIdx.x * blockDim.x + threadIdx.x;
    if (i < n) deg[i] = 1.0f;
}

// deg[dst[e]] += 1 over all edges
__global__ void gcn_count_deg(const int* __restrict__ dst, float* __restrict__ deg, int e) {
    int i = blockIdx.x * blockDim.x + threadIdx.x;
    if (i < e) atomicAdd(&deg[dst[i]], 1.0f);
}

// deg -> dinv = rsqrt(deg) in place (deg >= 1 always, no zero guard needed)
__global__ void gcn_rsqrt_deg(float* __restrict__ deg, int n) {
    int i = blockIdx.x * blockDim.x + threadIdx.x;
    if (i < n) deg[i] = rsqrtf(deg[i]);
}

// ---------------------------------------------------------------------------
// xw = x @ W using V_WMMA_F32_16X16X4_F32 (exact fp32 accumulate).
// Block = 256 threads = 8 waves; wave w computes the 16x16 tile at
// (m0 = 16*blockIdx.x, n0 = 16*w).  K = 128 -> 32 WMMA steps of K=4.
//
// A (16x4 f32) per ISA: VGPR0 = K+0 (lanes 0-15) / K+2 (lanes 16-31),
//                       VGPR1 = K+1 / K+3  -> one float2 load per lane.
// B (4x16 f32):         VGPR0 = row K+0 (lanes 0-15) / row K+2 (lanes 16-31),
//                       VGPR1 = row K+1 / K+3 -> two b32 loads per lane.
// C/D (16x16 f32): VGPR r = row r (lanes 0-15) / row r+8 (lanes 16-31).
__global__ __launch_bounds__(256) void gcn_gemm_wmma(const float* __restrict__ x,
                                                     const float* __restrict__ W,
                                                     float* __restrict__ xw,
                                                     int n_nodes) {
    const int lane = threadIdx.x & 31;
    const int wave = threadIdx.x >> 5;      // 0..7 -> N tile
    const int half = lane >> 4;             // 0: lanes 0-15, 1: lanes 16-31
    const int l    = lane & 15;

    const int m0 = blockIdx.x * 16;
    const int n0 = wave * 16;

    // Clamp A row so EXEC can stay all-1s even for a ragged last tile.
    int arow = m0 + l;
    if (arow >= n_nodes) arow = n_nodes - 1;

    const float* aptr = x + (size_t)arow * NFEAT + 2 * half;   // K offset 0 or 2
    const float* bptr = W + (size_t)(2 * half) * NHID + n0 + l; // row K+0 or K+2

    v8f c = {};
#pragma unroll
    for (int kk = 0; kk < NFEAT; kk += 4) {
        v2f a = *(const v2f*)(aptr + kk);                 // K+{0,1} or K+{2,3}
        v2f b;
        b.x = bptr[(size_t)kk * NHID];                    // row kk (+2*half)
        b.y = bptr[(size_t)kk * NHID + NHID];             // row kk+1 (+2*half)
        c = __builtin_amdgcn_wmma_f32_16x16x4_f32(
                /*neg_a=*/false, a, /*neg_b=*/false, b,
                /*c_mod=*/(short)0, c, /*reuse_a=*/false, /*reuse_b=*/false);
    }

#pragma unroll
    for (int r = 0; r < 8; ++r) {
        int row = m0 + r + 8 * half;
        if (row < n_nodes) xw[(size_t)row * NHID + n0 + l] = c[r];
    }
}

// ---------------------------------------------------------------------------
// out[i][:] = b + xw[i][:] * dinv[i]^2   (self-loop message + bias)
// one wave handles one node's 128 features: 4 floats per lane (float4).
__global__ void gcn_self_bias(const float* __restrict__ xw,
                              const float* __restrict__ dinv,
                              const float* __restrict__ bias,
                              float* __restrict__ out, int n_nodes) {
    int t = blockIdx.x * blockDim.x + threadIdx.x;
    int i = t >> 5;
    if (i >= n_nodes) return;
    int lane = t & 31;

    float di = dinv[i];
    float n2 = di * di;
    const float4 v  = *(const float4*)(xw + (size_t)i * NHID + lane * 4);
    const float4 bb = *(const float4*)(bias + lane * 4);
    float4 o;
    o.x = fmaf(v.x, n2, bb.x);
    o.y = fmaf(v.y, n2, bb.y);
    o.z = fmaf(v.z, n2, bb.z);
    o.w = fmaf(v.w, n2, bb.w);
    *(float4*)(out + (size_t)i * NHID + lane * 4) = o;
}

// ---------------------------------------------------------------------------
// One wave32 per edge: norm = dinv[src]*dinv[dst];
// lane grabs a coalesced float4 of xw[src] and does 4 f32 atomic adds to out[dst].
__global__ __launch_bounds__(256) void gcn_edge_scatter(const int* __restrict__ src,
                                                        const int* __restrict__ dst,
                                                        const float* __restrict__ dinv,
                                                        const float* __restrict__ xw,
                                                        float* __restrict__ out, int e) {
    int edge = blockIdx.x * 8 + (threadIdx.x >> 5);
    if (edge >= e) return;
    int lane = threadIdx.x & 31;

    int s = src[edge];
    int d = dst[edge];
    float norm = dinv[s] * dinv[d];

    const float4 v = *(const float4*)(xw + (size_t)s * NHID + lane * 4);
    float* o = out + (size_t)d * NHID + lane * 4;
    atomicAdd(o + 0, v.x * norm);
    atomicAdd(o + 1, v.y * norm);
    atomicAdd(o + 2, v.z * norm);
    atomicAdd(o + 3, v.w * norm);
}

// ---------------------------------------------------------------------------
extern "C" void kernel_launch(void* const* d_in, const int* in_sizes, int n_in,
                              void* d_out, int out_size, void* d_ws, size_t ws_size,
                              hipStream_t stream) {
    const int*   ei   = (const int*)d_in[0];   // edge_index [2, E]
    const float* x    = (const float*)d_in[1]; // [N, 128]
    const float* W    = (const float*)d_in[2]; // [128, 128]
    const float* bias = (const float*)d_in[3]; // [128]
    float*       out  = (float*)d_out;         // [N, 128]

    const int E = in_sizes[0] / 2;
    const int N = in_sizes[1] / NFEAT;

    // workspace layout: xw [N*128 f32] | deg/dinv [N f32]
    float* xw  = (float*)d_ws;
    float* deg = (float*)((char*)d_ws + (size_t)N * NHID * sizeof(float));

    const int* srcIdx = ei;
    const int* dstIdx = ei + E;

    gcn_init_deg <<<(N + 255) / 256, 256, 0, stream>>>(deg, N);
    gcn_count_deg<<<(E + 255) / 256, 256, 0, stream>>>(dstIdx, deg, E);
    gcn_rsqrt_deg<<<(N + 255) / 256, 256, 0, stream>>>(deg, N);

    gcn_gemm_wmma<<<(N + 15) / 16, 256, 0, stream>>>(x, W, xw, N);

    gcn_self_bias<<<((size_t)N * 32 + 255) / 256, 256, 0, stream>>>(xw, deg, bias, out, N);
    gcn_edge_scatter<<<(E + 7) / 8, 256, 0, stream>>>(srcIdx, dstIdx, deg, xw, out, E);
}